// SpanScorerCRF_14439680049700
// MI455X (gfx1250) — compile-verified
//
#include <hip/hip_runtime.h>

#define B_ 64
#define S_ 512
#define E_ 1024
#define T_ 9
#define TP 16               // padded T (N dimension of WMMA tile)
#define KCHUNK 64
#define NCHUNKS (E_ / KCHUNK)
#define LDS_STRIDE 68       // 64 data + 4 pad DWORDs -> 16 row reads hit 16 distinct banks

typedef float v2f __attribute__((ext_vector_type(2)));
typedef float v8f __attribute__((ext_vector_type(8)));
typedef unsigned int v4u __attribute__((ext_vector_type(4)));
typedef int v8i __attribute__((ext_vector_type(8)));
typedef int v4i __attribute__((ext_vector_type(4)));

// ---------------------------------------------------------------- pad W ----
__global__ void pad_w_kernel(const float* __restrict__ W, float* __restrict__ Wp) {
  int i = blockIdx.x * blockDim.x + threadIdx.x;     // 0 .. E_*TP-1
  if (i < E_ * TP) {
    int e = i >> 4, t = i & 15;
    Wp[i] = (t < T_) ? W[e * T_ + t] : 0.0f;
  }
}

// ------------------------------------- logits GEMM: TDM double-buffer + WMMA
// One wave per 16-row M tile. A tiles (16x64 fp32) DMA'd global->LDS by the
// Tensor Data Mover with LDS padding (row stride 68 DWORDs), double buffered
// on TENSORcnt; 256 x V_WMMA_F32_16X16X4_F32 per wave. No block barriers.
__global__ void logits_wmma_kernel(const float* __restrict__ X,
                                   const float* __restrict__ Wp,
                                   const float* __restrict__ bias,
                                   float* __restrict__ logits) {
  __shared__ float tile[8][2][16 * LDS_STRIDE];      // 8 waves x double buffer
  const int lane = threadIdx.x & 31;
  const int wave = threadIdx.x >> 5;
  const int mtile = blockIdx.x * 8 + wave;           // 0 .. 2047
  const long row0 = (long)mtile * 16;

  const unsigned long long gaddr0 =
      (unsigned long long)(const void*)(X + row0 * E_);
  const unsigned int lds0 =
      (unsigned int)(unsigned long long)(void*)&tile[wave][0][0];
  const unsigned int lds1 =
      (unsigned int)(unsigned long long)(void*)&tile[wave][1][0];

  // ---- D# group1 (constant per wave): data_size=4B, pad 4 DW every 64 DW,
  //      tensor dims huge (no OOB), tile 64x16, dim0_stride = E_ elements.
  v8i g1;
  const unsigned int td0 = 1u << 20, td1 = 1u << 20;
  g1[0] = (2 << 16) | (1 << 20) | (5 << 22) | (3 << 25);
  g1[1] = (int)((td0 & 0xFFFFu) << 16);                   // tensor_dim0 lo16
  g1[2] = (int)((td0 >> 16) | ((td1 & 0xFFFFu) << 16));   // dim0 hi / dim1 lo
  g1[3] = (int)((td1 >> 16) | (64u << 16));               // dim1 hi / tile_dim0=64
  g1[4] = 16;                                             // tile_dim1=16, tile_dim2=0
  g1[5] = E_;                                             // tensor_dim0_stride lo32
  g1[6] = 0;
  g1[7] = 0;
  const v4i gz4 = {0, 0, 0, 0};
  const v8i gz8 = {0, 0, 0, 0, 0, 0, 0, 0};

  auto issue = [&](unsigned int ldsAddr, int k) {
    const unsigned long long ga = gaddr0 + (unsigned long long)k * 4ull;
    v4u g0;
    g0[0] = 1u;                                           // count=1 (valid)
    g0[1] = ldsAddr;                                      // lds_addr
    g0[2] = (unsigned int)(ga & 0xFFFFFFFFu);             // global_addr lo
    g0[3] = (unsigned int)((ga >> 32) & 0x1FFFFFFu) | (2u << 30);  // hi | type=2
    __builtin_amdgcn_tensor_load_to_lds(g0, g1, gz4, gz4, gz8, 0);
  };

  v8f acc = {};
  const int m  = lane & 15;                          // A fragment row
  const int kb = (lane >> 4) * 2;                    // A/B fragment K sub-offset
  const int n  = lane & 15;                          // B/C fragment column

  issue(lds0, 0);
  int cur = 0;
  for (int kc = 0; kc < NCHUNKS; ++kc) {
    if (kc + 1 < NCHUNKS) {
      issue(cur ? lds0 : lds1, (kc + 1) * KCHUNK);
      __builtin_amdgcn_s_wait_tensorcnt((short)1);   // previous tile landed
    } else {
      __builtin_amdgcn_s_wait_tensorcnt((short)0);
    }
    asm volatile("" ::: "memory");                   // LDS written by TDM
    const float* t = cur ? &tile[wave][1][0] : &tile[wave][0][0];
    const int kbase = kc * KCHUNK;
#pragma unroll
    for (int k4 = 0; k4 < KCHUNK; k4 += 4) {
      v2f a, bf;
      a.x = t[m * LDS_STRIDE + k4 + kb];
      a.y = t[m * LDS_STRIDE + k4 + kb + 1];
      const float* wp = Wp + (long)(kbase + k4 + kb) * TP + n;
      bf.x = wp[0];
      bf.y = wp[TP];
      acc = __builtin_amdgcn_wmma_f32_16x16x4_f32(false, a, false, bf,
                                                  (short)0, acc, false, false);
    }
    cur ^= 1;
  }

  // C layout: VGPR j -> M = j + 8*(lane>=16), N = lane&15
  if (n < T_) {
    const float bn = bias[n];
    const int mofs = (lane >> 4) * 8;
#pragma unroll
    for (int j = 0; j < 8; ++j) {
      long row = row0 + mofs + j;
      logits[row * T_ + n] = acc[j] + bn;
    }
  }
}

// ----------------------------------------- CRF forward (logZ) + gold score --
__global__ void crf_forward_kernel(const float* __restrict__ logits,
                                   const unsigned char* __restrict__ mask,
                                   const int* __restrict__ labels,
                                   const float* __restrict__ trans,
                                   const float* __restrict__ start_t,
                                   const float* __restrict__ end_t,
                                   float* __restrict__ part) {
  const int b = blockIdx.x;
  const int lane = threadIdx.x;
  __shared__ float alphaS[2][TP];
  __shared__ float red[32];
  __shared__ float redc[32];

  const float* lg = logits + (long)b * S_ * T_;
  const unsigned char* mk = mask + (long)b * S_;
  const int* tg = labels + (long)b * S_;

  if (lane < T_) alphaS[0][lane] = start_t[lane] + lg[lane];
  __syncthreads();
  int cur = 0;
  for (int s = 1; s < S_; ++s) {
    float newv = 0.f;
    if (lane < T_) {
      float mx = -3.402823466e38f;
#pragma unroll
      for (int i = 0; i < T_; ++i)
        mx = fmaxf(mx, alphaS[cur][i] + trans[i * T_ + lane]);
      float sum = 0.f;
#pragma unroll
      for (int i = 0; i < T_; ++i)
        sum += __expf(alphaS[cur][i] + trans[i * T_ + lane] - mx);
      newv = mx + __logf(sum) + lg[s * T_ + lane];
      if (!mk[s]) newv = alphaS[cur][lane];
    }
    __syncthreads();
    if (lane < T_) alphaS[cur ^ 1][lane] = newv;
    cur ^= 1;
    __syncthreads();
  }

  float acc = 0.f, cnt = 0.f;
  for (int s = lane; s < S_; s += 32) {
    float mf = mk[s] ? 1.f : 0.f;
    cnt += mf;
    if (s < S_ - 1) acc += lg[s * T_ + tg[s]] * mf;
    if (s >= 1)     acc += trans[tg[s - 1] * T_ + tg[s]] * mf;
  }
  red[lane] = acc;
  redc[lane] = cnt;
  __syncthreads();
  if (lane == 0) {
    float gold = 0.f, c = 0.f;
    for (int i = 0; i < 32; ++i) { gold += red[i]; c += redc[i]; }
    int last_idx = (int)c - 1;
    if (last_idx < 0) last_idx = 0;
    int last_tag = tg[last_idx];
    gold += start_t[tg[0]] + end_t[last_tag];
    gold += lg[(S_ - 1) * T_ + last_tag] * (mk[S_ - 1] ? 1.f : 0.f);
    float mx = -3.402823466e38f;
    for (int t = 0; t < T_; ++t) mx = fmaxf(mx, alphaS[cur][t] + end_t[t]);
    float sum = 0.f;
    for (int t = 0; t < T_; ++t) sum += __expf(alphaS[cur][t] + end_t[t] - mx);
    float logZ = mx + __logf(sum);
    part[b] = gold - logZ;
  }
}

// ---------------------------------------------------------- loss reduce ----
__global__ void loss_reduce_kernel(const float* __restrict__ part,
                                   float* __restrict__ out) {
  __shared__ float red[B_];
  int t = threadIdx.x;
  red[t] = part[t];
  __syncthreads();
  if (t == 0) {
    float s = 0.f;
    for (int i = 0; i < B_; ++i) s += red[i];
    out[0] = -s;
  }
}

// ------------------------------------------------------------- Viterbi ----
__global__ void viterbi_kernel(const float* __restrict__ logits,
                               const unsigned char* __restrict__ mask,
                               const float* __restrict__ trans,
                               const float* __restrict__ start_t,
                               const float* __restrict__ end_t,
                               int* __restrict__ bp,
                               float* __restrict__ out) {
  const int b = blockIdx.x;
  const int lane = threadIdx.x;
  __shared__ float dS[2][TP];
  const float* lg = logits + (long)b * S_ * T_;
  const unsigned char* mk = mask + (long)b * S_;
  int* bpb = bp + (long)b * S_ * T_;

  if (lane < T_) dS[0][lane] = start_t[lane] + lg[lane];
  __syncthreads();
  int cur = 0;
  for (int s = 1; s < S_; ++s) {
    float newv = 0.f;
    if (lane < T_) {
      float best = -3.402823466e38f; int arg = 0;
#pragma unroll
      for (int i = 0; i < T_; ++i) {
        float v = dS[cur][i] + trans[i * T_ + lane];
        if (v > best) { best = v; arg = i; }          // first-max = jnp.argmax
      }
      newv = best + lg[s * T_ + lane];
      if (!mk[s]) { newv = dS[cur][lane]; arg = lane; }
      bpb[s * T_ + lane] = arg;
    }
    __syncthreads();
    if (lane < T_) dS[cur ^ 1][lane] = newv;
    cur ^= 1;
    __syncthreads();
  }
  if (lane == 0) {
    float best = -3.402823466e38f; int arg = 0;
    for (int t = 0; t < T_; ++t) {
      float v = dS[cur][t] + end_t[t];
      if (v > best) { best = v; arg = t; }
    }
    out[1 + B_ * S_ + b] = best;                      // vit_score
    int tag = arg;
    float* pred = out + 1 + (long)b * S_;
    pred[S_ - 1] = (float)tag;
    for (int s = S_ - 1; s >= 1; --s) {
      tag = bpb[s * T_ + tag];
      pred[s - 1] = (float)tag;
    }
  }
}

// -------------------------------------------------------------- launch ----
extern "C" void kernel_launch(void* const* d_in, const int* in_sizes, int n_in,
                              void* d_out, int out_size, void* d_ws, size_t ws_size,
                              hipStream_t stream) {
  (void)in_sizes; (void)n_in; (void)out_size; (void)ws_size;
  const float*         seq    = (const float*)d_in[0];
  const unsigned char* mask   = (const unsigned char*)d_in[1];  // jnp bool = 1 byte
  const int*           labels = (const int*)d_in[2];
  const float*         W      = (const float*)d_in[3];
  const float*         bias   = (const float*)d_in[4];
  const float*         trans  = (const float*)d_in[5];
  const float*         startt = (const float*)d_in[6];
  const float*         endt   = (const float*)d_in[7];
  float* out = (float*)d_out;

  char* ws = (char*)d_ws;
  float* Wp     = (float*)ws;                              // 1024*16*4   = 64 KB
  float* logits = (float*)(ws + 65536);                    // 32768*9*4   = 1,179,648 B
  float* part   = (float*)(ws + 65536 + 1179648);          // 64*4        = 256 B
  int*   bp     = (int*)  (ws + 65536 + 1179648 + 256);    // 64*512*9*4  = 1,179,648 B

  pad_w_kernel      <<<64, 256, 0, stream>>>(W, Wp);
  logits_wmma_kernel<<<256, 256, 0, stream>>>(seq, Wp, bias, logits);
  crf_forward_kernel<<<B_, 32, 0, stream>>>(logits, mask, labels, trans, startt, endt, part);
  loss_reduce_kernel<<<1, B_, 0, stream>>>(part, out);
  viterbi_kernel    <<<B_, 32, 0, stream>>>(logits, mask, trans, startt, endt, bp, out);
}